// MaxPoolAggregator_38405597561096
// MI455X (gfx1250) — compile-verified
//
#include <hip/hip_runtime.h>

// ---------------------------------------------------------------------------
// MaxPoolAggregator for MI455X / gfx1250 (wave32, WMMA)
//
// Phase 1: h[t][n][o] = sum_f x[n][f][t] * W[o][f] + b[o]
//          via V_WMMA_F32_16X16X4_F32 (fp32 tensor op, exact precision).
// Phase 2: init agg[t][n][o] = -inf
// Phase 3: per-edge scatter: agg[t][dst][:] = max(agg, h[t][src][:])
//          using native global_atomic_max_num_f32 (no-return, device scope).
//          h (41MB) + agg (41MB) fit in the 192MB L2 -> L2-resident phase.
// Phase 4: out[n][o][t] = (agg[t][n][o] == -inf) ? 0 : agg[t][n][o]
// ---------------------------------------------------------------------------

typedef __attribute__((ext_vector_type(2))) float v2f;
typedef __attribute__((ext_vector_type(8))) float v8f;

#define N_NODES 20000
#define N_EDGES 640000
#define FEAT    128
#define T_DIM   4
#define NF      (N_NODES * FEAT)       // one t-plane, 2,560,000 floats

// ---------------- Phase 1: fp32 WMMA GEMM ----------------------------------
// One wave computes one 16-node tile x all 128 output features for one t.
// A (16x4 f32): lanes 0-15 -> M=lane, K={0,1} in the two VGPRs;
//               lanes 16-31 -> M=lane-16, K={2,3}.
// B (4x16 f32): same pattern with N instead of M.  B[k][n] = W[n][k].
// C/D (16x16 f32): lanes 0-15 N=lane, M=vgpr(0..7); lanes 16-31 N=lane-16, M=8+vgpr.
__global__ __launch_bounds__(256) void gemm_wmma_f32(
    const float* __restrict__ x,     // [N_NODES][FEAT][T_DIM]
    const float* __restrict__ W,     // [FEAT][FEAT]  (o-major)
    const float* __restrict__ bias,  // [FEAT]
    float* __restrict__ h)           // [T_DIM][N_NODES][FEAT]
{
    const int lane  = threadIdx.x & 31;
    const int wave  = threadIdx.x >> 5;
    const int l     = lane & 15;
    const int half  = lane >> 4;        // 0: K pair {0,1}; 1: K pair {2,3}
    const int t     = blockIdx.y;
    const int m_tile = (blockIdx.x * 8 + wave) * 16;
    if (m_tile >= N_NODES) return;      // wave-uniform; EXEC stays all-1s

    v8f acc[8];
#pragma unroll
    for (int i = 0; i < 8; ++i) acc[i] = (v8f)(0.0f);

    // A-fragment source row for this lane (x[n][f][t], f-stride = T_DIM floats)
    const float* xrow = x + (size_t)(m_tile + l) * (FEAT * T_DIM) + t;

    for (int kb = 0; kb < FEAT / 4; ++kb) {
        const int k0 = kb * 4 + half * 2;
        v2f a;
        a.x = xrow[(k0 + 0) * T_DIM];
        a.y = xrow[(k0 + 1) * T_DIM];
#pragma unroll
        for (int nt = 0; nt < 8; ++nt) {          // all 8 n-tiles reuse A regs
            const float* wrow = W + (size_t)(nt * 16 + l) * FEAT + k0;
            v2f bm;
            bm.x = wrow[0];
            bm.y = wrow[1];
            acc[nt] = __builtin_amdgcn_wmma_f32_16x16x4_f32(
                false, a, false, bm, (short)0, acc[nt], false, false);
        }
    }

    // Store D + bias into h[t][m][o]
#pragma unroll
    for (int nt = 0; nt < 8; ++nt) {
        const float bv = bias[nt * 16 + l];
        float* hp = h + (size_t)t * NF
                      + (size_t)(m_tile + 8 * half) * FEAT + nt * 16 + l;
#pragma unroll
        for (int r = 0; r < 8; ++r)
            hp[(size_t)r * FEAT] = acc[nt][r] + bv;
    }
}

// ---------------- Phase 2: init aggregation buffer to -inf -----------------
__global__ __launch_bounds__(256) void init_neg_inf(float4* __restrict__ agg, int n4)
{
    const int i = blockIdx.x * blockDim.x + threadIdx.x;
    if (i < n4) {
        const float ninf = -__builtin_inff();
        agg[i] = make_float4(ninf, ninf, ninf, ninf);
    }
}

// ---------------- Phase 3: edge scatter with native f32 max atomics --------
__device__ __forceinline__ void atomic_max_f32(float* p, float v)
{
    // No-return IEEE maximumNumber atomic; tracked on STOREcnt, device scope.
    asm volatile("global_atomic_max_num_f32 %0, %1, off scope:SCOPE_DEV"
                 :: "v"((unsigned long long)p), "v"(v)
                 : "memory");
}

__global__ __launch_bounds__(256) void scatter_max(
    const int* __restrict__ ei,      // [2][N_EDGES]  (row0=src, row1=dst)
    const float* __restrict__ h,     // [T_DIM][N_NODES][FEAT]
    float* __restrict__ agg)         // [T_DIM][N_NODES][FEAT]
{
    const int lane = threadIdx.x & 31;
    const int wave = threadIdx.x >> 5;
    const int e = blockIdx.x * 8 + wave;     // one wave per edge
    if (e >= N_EDGES) return;

    const int src = ei[e];
    const int dst = ei[N_EDGES + e];

#pragma unroll
    for (int t = 0; t < T_DIM; ++t) {
        // 32 lanes x float4 = 128 contiguous floats (one B128 load per lane)
        const float4 v = *(const float4*)(h + (size_t)t * NF
                                            + (size_t)src * FEAT + lane * 4);
        float* base = agg + (size_t)t * NF + (size_t)dst * FEAT + lane * 4;
        atomic_max_f32(base + 0, v.x);
        atomic_max_f32(base + 1, v.y);
        atomic_max_f32(base + 2, v.z);
        atomic_max_f32(base + 3, v.w);
    }
}

// ---------------- Phase 4: -inf -> 0 and transpose to (n,o,t) --------------
__global__ __launch_bounds__(256) void finalize_transpose(
    const float* __restrict__ agg,   // [T_DIM][N_NODES][FEAT]
    float4* __restrict__ out)        // [N_NODES][FEAT] of float4 over t
{
    const int i = blockIdx.x * blockDim.x + threadIdx.x;   // i = n*FEAT + o
    if (i >= NF) return;
    const float ninf = -__builtin_inff();
    float v0 = agg[0 * (size_t)NF + i];
    float v1 = agg[1 * (size_t)NF + i];
    float v2 = agg[2 * (size_t)NF + i];
    float v3 = agg[3 * (size_t)NF + i];
    float4 o4;
    o4.x = (v0 == ninf) ? 0.0f : v0;
    o4.y = (v1 == ninf) ? 0.0f : v1;
    o4.z = (v2 == ninf) ? 0.0f : v2;
    o4.w = (v3 == ninf) ? 0.0f : v3;
    out[i] = o4;                      // coalesced 16B stores
}

// ---------------------------------------------------------------------------
extern "C" void kernel_launch(void* const* d_in, const int* in_sizes, int n_in,
                              void* d_out, int out_size, void* d_ws, size_t ws_size,
                              hipStream_t stream)
{
    const float* x    = (const float*)d_in[0];   // (1, N, F, T) fp32
    const int*   ei   = (const int*)d_in[1];     // (2, E) integer
    const float* W    = (const float*)d_in[2];   // (F, F) fp32
    const float* bias = (const float*)d_in[3];   // (F,)  fp32

    float* h   = (float*)d_ws;                   // [T][N][F]  40.96 MB
    float* agg = h + (size_t)T_DIM * NF;         // [T][N][F]  40.96 MB
    float* out = (float*)d_out;                  // [N][F][T]

    // Phase 1: projection GEMM (8 waves/block, 16 nodes per wave, t on grid.y)
    dim3 gemm_grid((N_NODES / 16 + 7) / 8, T_DIM);
    gemm_wmma_f32<<<gemm_grid, 256, 0, stream>>>(x, W, bias, h);

    // Phase 2: agg = -inf
    const int n4 = (T_DIM * NF) / 4;
    init_neg_inf<<<(n4 + 255) / 256, 256, 0, stream>>>((float4*)agg, n4);

    // Phase 3: edge scatter-max (one wave per edge; L2-resident h/agg)
    scatter_max<<<N_EDGES / 8, 256, 0, stream>>>(ei, h, agg);

    // Phase 4: fixup + transpose into output layout
    finalize_transpose<<<(NF + 255) / 256, 256, 0, stream>>>(agg, (float4*)out);

    (void)in_sizes; (void)n_in; (void)out_size; (void)ws_size;
}